// SPGPromptEncoder_29205777613580
// MI455X (gfx1250) — compile-verified
//
#include <hip/hip_runtime.h>
#include <math.h>

#define H_IN 1024
#define W_IN 1024
#define NPIX (H_IN * W_IN)          // 1048576
#define BATCH 8
#define CHUNKS 64                    // reduce blocks per batch
#define PIX_PER_THREAD 64            // 256 thr * 64 px * 64 blk = 1M px

typedef float v2f __attribute__((ext_vector_type(2)));
typedef float v8f __attribute__((ext_vector_type(8)));

__device__ __forceinline__ float gelu_exact(float x) {
    return 0.5f * x * (1.0f + erff(x * 0.70710678118654752440f));
}

// ---------------- K1: diff = mean_c |img - mir| ; per-block S1,S2 ----------------
__global__ void k1_diff_stats(const float* __restrict__ img, const float* __restrict__ mir,
                              float* __restrict__ diff, double* __restrict__ partials) {
    int b = blockIdx.y, chunk = blockIdx.x, t = threadIdx.x;
    const float* ib = img + (size_t)b * 3 * NPIX;
    const float* mb = mir + (size_t)b * 3 * NPIX;
    float* db = diff + (size_t)b * NPIX;
    double s1 = 0.0, s2 = 0.0;
    for (int i = 0; i < PIX_PER_THREAD; ++i) {
        int p = chunk * 16384 + i * 256 + t;           // coalesced
        float d0 = fabsf(ib[p]            - mb[p]);
        float d1 = fabsf(ib[p +   NPIX]   - mb[p +   NPIX]);
        float d2 = fabsf(ib[p + 2*NPIX]   - mb[p + 2*NPIX]);
        float d  = (d0 + d1 + d2) * (1.0f / 3.0f);
        db[p] = d;
        s1 += (double)d;
        s2 += (double)d * (double)d;
    }
    __shared__ double sh1[256], sh2[256];
    sh1[t] = s1; sh2[t] = s2;
    __syncthreads();
    for (int s = 128; s > 0; s >>= 1) {
        if (t < s) { sh1[t] += sh1[t + s]; sh2[t] += sh2[t + s]; }
        __syncthreads();
    }
    if (t == 0) {
        partials[(b * CHUNKS + chunk) * 2 + 0] = sh1[0];
        partials[(b * CHUNKS + chunk) * 2 + 1] = sh2[0];
    }
}

// ---------------- K2: thresh[b] = mean + std(ddof=1) ----------------
__global__ void k2_thresh(const double* __restrict__ partials, float* __restrict__ thresh) {
    int b = blockIdx.x, t = threadIdx.x;               // 64 threads
    __shared__ double s1[64], s2[64];
    s1[t] = partials[(b * CHUNKS + t) * 2 + 0];
    s2[t] = partials[(b * CHUNKS + t) * 2 + 1];
    __syncthreads();
    for (int s = 32; s > 0; s >>= 1) {
        if (t < s) { s1[t] += s1[t + s]; s2[t] += s2[t + s]; }
        __syncthreads();
    }
    if (t == 0) {
        double N = (double)NPIX;
        double mean = s1[0] / N;
        double var = (s2[0] - s1[0] * s1[0] / N) / (N - 1.0);
        if (var < 0.0) var = 0.0;
        thresh[b] = (float)(mean + sqrt(var));
    }
}

// ---------------- K3: centroid partials from transient mask ----------------
__global__ void k3_centroid(const float* __restrict__ diff, const float* __restrict__ thresh,
                            double* __restrict__ partials2) {
    int b = blockIdx.y, chunk = blockIdx.x, t = threadIdx.x;
    const float* db = diff + (size_t)b * NPIX;
    float th = thresh[b];
    double tot = 0.0, sr = 0.0, sc = 0.0;
    for (int i = 0; i < PIX_PER_THREAD; ++i) {
        int p = chunk * 16384 + i * 256 + t;
        if (db[p] > th) {
            tot += 1.0;
            sr += (double)(p >> 10);     // row
            sc += (double)(p & 1023);    // col
        }
    }
    __shared__ double st[256], ssr[256], ssc[256];
    st[t] = tot; ssr[t] = sr; ssc[t] = sc;
    __syncthreads();
    for (int s = 128; s > 0; s >>= 1) {
        if (t < s) { st[t] += st[t + s]; ssr[t] += ssr[t + s]; ssc[t] += ssc[t + s]; }
        __syncthreads();
    }
    if (t == 0) {
        partials2[(b * CHUNKS + chunk) * 3 + 0] = st[0];
        partials2[(b * CHUNKS + chunk) * 3 + 1] = ssr[0];
        partials2[(b * CHUNKS + chunk) * 3 + 2] = ssc[0];
    }
}

// ---------------- K4: finalize centroid, Gaussian PE, sparse output ----------------
__global__ void k4_sparse(const double* __restrict__ partials2,
                          const float* __restrict__ pe_gauss,
                          const float* __restrict__ pos_pe,
                          const float* __restrict__ nap,
                          float* __restrict__ sparse) {
    int b = blockIdx.x, t = threadIdx.x;               // 128 threads
    __shared__ double st[64], sr[64], sc[64];
    if (t < 64) {
        st[t] = partials2[(b * CHUNKS + t) * 3 + 0];
        sr[t] = partials2[(b * CHUNKS + t) * 3 + 1];
        sc[t] = partials2[(b * CHUNKS + t) * 3 + 2];
    }
    __syncthreads();
    for (int s = 32; s > 0; s >>= 1) {
        if (t < s) { st[t] += st[t + s]; sr[t] += sr[t + s]; sc[t] += sc[t + s]; }
        __syncthreads();
    }
    __shared__ float cxy[2];
    if (t == 0) {
        double tot  = st[0];
        double safe = tot > 1.0 ? tot : 1.0;
        cxy[0] = (tot > 0.0) ? (float)(sc[0] / safe) : (W_IN * 0.5f);  // cx
        cxy[1] = (tot > 0.0) ? (float)(sr[0] / safe) : (H_IN * 0.5f);  // cy
    }
    __syncthreads();
    float c0 = 2.0f * ((cxy[0] + 0.5f) / (float)W_IN) - 1.0f;
    float c1 = 2.0f * ((cxy[1] + 0.5f) / (float)H_IN) - 1.0f;
    float proj = 6.283185307179586f * (c0 * pe_gauss[t] + c1 * pe_gauss[128 + t]);
    float* row = sparse + (size_t)b * 512;             // (2,256) per batch
    row[t]        = sinf(proj) + pos_pe[t];
    row[128 + t]  = cosf(proj) + pos_pe[128 + t];
    row[256 + t]  = nap[t];
    row[384 + t]  = nap[128 + t];
}

// ---------------- K5: conv1(2x2,s2) 1->4 + LN(4) + GELU, pixel-major out ----------------
__global__ void k5_conv1(const float* __restrict__ diff, const float* __restrict__ thresh,
                         const float* __restrict__ w1, const float* __restrict__ b1,
                         const float* __restrict__ ga1, const float* __restrict__ be1,
                         float* __restrict__ x1) {
    int idx = blockIdx.x * blockDim.x + threadIdx.x;   // B*512*512
    int b = idx >> 18;
    int p = idx & 262143;
    int oy = p >> 9, ox = p & 511;
    const float* db = diff + (size_t)b * NPIX;
    float th = thresh[b];
    float m00 = db[(size_t)(2 * oy) * W_IN + 2 * ox]         > th ? 1.f : 0.f;
    float m01 = db[(size_t)(2 * oy) * W_IN + 2 * ox + 1]     > th ? 1.f : 0.f;
    float m10 = db[(size_t)(2 * oy + 1) * W_IN + 2 * ox]     > th ? 1.f : 0.f;
    float m11 = db[(size_t)(2 * oy + 1) * W_IN + 2 * ox + 1] > th ? 1.f : 0.f;
    float v[4];
    float mean = 0.f;
    #pragma unroll
    for (int c = 0; c < 4; ++c) {
        float a = b1[c] + w1[c*4+0]*m00 + w1[c*4+1]*m01 + w1[c*4+2]*m10 + w1[c*4+3]*m11;
        v[c] = a; mean += a;
    }
    mean *= 0.25f;
    float var = 0.f;
    #pragma unroll
    for (int c = 0; c < 4; ++c) { float d = v[c] - mean; var += d * d; }
    var *= 0.25f;
    float inv = rsqrtf(var + 1e-6f);
    float4 o;
    o.x = gelu_exact((v[0] - mean) * inv * ga1[0] + be1[0]);
    o.y = gelu_exact((v[1] - mean) * inv * ga1[1] + be1[1]);
    o.z = gelu_exact((v[2] - mean) * inv * ga1[2] + be1[2]);
    o.w = gelu_exact((v[3] - mean) * inv * ga1[3] + be1[3]);
    *(float4*)(x1 + (size_t)idx * 4) = o;
}

// ---------------- K6: conv2(2x2,s2) 4->16 + LN(16) + GELU, pixel-major out ----------------
__global__ void k6_conv2(const float* __restrict__ x1,
                         const float* __restrict__ w2, const float* __restrict__ b2,
                         const float* __restrict__ ga2, const float* __restrict__ be2,
                         float* __restrict__ x2) {
    __shared__ float ws2[256];                          // (16,4,2,2)
    ws2[threadIdx.x] = w2[threadIdx.x];
    __syncthreads();
    int idx = blockIdx.x * blockDim.x + threadIdx.x;    // B*256*256
    int b = idx >> 16;
    int p = idx & 65535;
    int oy = p >> 8, ox = p & 255;
    const float* xb = x1 + (size_t)b * (512 * 512 * 4);
    float in[2][2][4];
    #pragma unroll
    for (int dy = 0; dy < 2; ++dy)
        #pragma unroll
        for (int dx = 0; dx < 2; ++dx) {
            float4 f = *(const float4*)(xb + ((size_t)(2*oy+dy)*512 + (2*ox+dx)) * 4);
            in[dy][dx][0] = f.x; in[dy][dx][1] = f.y;
            in[dy][dx][2] = f.z; in[dy][dx][3] = f.w;
        }
    float v[16]; float mean = 0.f;
    #pragma unroll
    for (int co = 0; co < 16; ++co) {
        float a = b2[co];
        #pragma unroll
        for (int ci = 0; ci < 4; ++ci)
            #pragma unroll
            for (int dy = 0; dy < 2; ++dy)
                #pragma unroll
                for (int dx = 0; dx < 2; ++dx)
                    a += ws2[((co*4 + ci)*2 + dy)*2 + dx] * in[dy][dx][ci];
        v[co] = a; mean += a;
    }
    mean *= (1.0f / 16.0f);
    float var = 0.f;
    #pragma unroll
    for (int co = 0; co < 16; ++co) { float d = v[co] - mean; var += d * d; }
    var *= (1.0f / 16.0f);
    float inv = rsqrtf(var + 1e-6f);
    float* out = x2 + (size_t)idx * 16;                 // K-contiguous for WMMA B
    #pragma unroll
    for (int co = 0; co < 16; ++co)
        out[co] = gelu_exact((v[co] - mean) * inv * ga2[co] + be2[co]);
}

// ---------------- K7: conv3 1x1 16->256 as f32 WMMA GEMM  D = W * X^T ----------------
// Per wave: one 16-pixel strip x 256 channels. A = w3 tile (M=co, K=ci),
// B = x2^T tile (K=ci, N=px). K=16 done as 4x V_WMMA_F32_16X16X4_F32.
__global__ void k7_conv3_wmma(const float* __restrict__ x2,
                              const float* __restrict__ w3, const float* __restrict__ b3,
                              float* __restrict__ dense) {
    int lane = threadIdx.x & 31;
    int wave = threadIdx.x >> 5;
    int wt = blockIdx.x * 8 + wave;                     // wave tile id
    int pixBase = wt * 16;                              // global pixel across batches
    int n = lane & 15;                                  // N (pixel) / M (co row) index
    int khalf = (lane < 16) ? 0 : 2;                    // K sub-pair per lane half
    int coOff = (lane < 16) ? 0 : 8;                    // D rows M=r / M=r+8

    // Load B operand (X^T) once; reused across all 16 channel tiles.
    v2f bt[4];
    const float* xp = x2 + (size_t)(pixBase + n) * 16;
    #pragma unroll
    for (int kk = 0; kk < 4; ++kk)
        bt[kk] = *(const v2f*)(xp + kk * 4 + khalf);

    int bimg = pixBase >> 16;                           // batch
    int pin  = pixBase & 65535;                         // pixel within image
    float* dbase = dense + ((size_t)bimg << 24);        // b * 256 * 65536

    #pragma unroll 1
    for (int nt = 0; nt < 16; ++nt) {
        int coBase = nt * 16;
        v8f acc;
        #pragma unroll
        for (int r = 0; r < 8; ++r)
            acc[r] = b3[coBase + coOff + r];            // bias per D row
        #pragma unroll
        for (int kk = 0; kk < 4; ++kk) {
            v2f at = *(const v2f*)(w3 + (size_t)(coBase + n) * 16 + kk * 4 + khalf);
            acc = __builtin_amdgcn_wmma_f32_16x16x4_f32(
                false, at, false, bt[kk], (short)0, acc, false, false);
        }
        // D tile: row M = r + coOff (channel), col N = lane&15 (pixel) ->
        // lanes 0-15 / 16-31 each write 16 consecutive floats of one channel row.
        #pragma unroll
        for (int r = 0; r < 8; ++r)
            dbase[(size_t)(coBase + coOff + r) * 65536 + pin + n] = acc[r];
    }
}

extern "C" void kernel_launch(void* const* d_in, const int* in_sizes, int n_in,
                              void* d_out, int out_size, void* d_ws, size_t ws_size,
                              hipStream_t stream) {
    const float* image    = (const float*)d_in[0];
    const float* mirrored = (const float*)d_in[1];
    const float* pe_gauss = (const float*)d_in[2];
    const float* pos_pe   = (const float*)d_in[3];
    const float* nap      = (const float*)d_in[4];
    const float* w1  = (const float*)d_in[5];
    const float* b1  = (const float*)d_in[6];
    const float* ga1 = (const float*)d_in[7];
    const float* be1 = (const float*)d_in[8];
    const float* w2  = (const float*)d_in[9];
    const float* b2  = (const float*)d_in[10];
    const float* ga2 = (const float*)d_in[11];
    const float* be2 = (const float*)d_in[12];
    const float* w3  = (const float*)d_in[13];
    const float* b3  = (const float*)d_in[14];

    char* ws = (char*)d_ws;
    // diff is dead after K5; x2 (written in K6) aliases it.
    float*  diff      = (float*)(ws);                           // 33,554,432 B
    float*  x2        = (float*)(ws);                           // alias of diff
    float*  x1        = (float*)(ws + 33554432);                // 33,554,432 B
    double* partials  = (double*)(ws + 67108864);               //  8,192 B
    double* partials2 = (double*)(ws + 67108864 + 8192);        // 12,288 B
    float*  thresh    = (float*)(ws + 67108864 + 8192 + 12288); //     32 B

    float* sparse = (float*)d_out;          // (8,2,256) = 4096 floats
    float* dense  = (float*)d_out + 4096;   // (8,256,256,256)

    dim3 gred(CHUNKS, BATCH);
    k1_diff_stats<<<gred, 256, 0, stream>>>(image, mirrored, diff, partials);
    k2_thresh<<<BATCH, 64, 0, stream>>>(partials, thresh);
    k3_centroid<<<gred, 256, 0, stream>>>(diff, thresh, partials2);
    k4_sparse<<<BATCH, 128, 0, stream>>>(partials2, pe_gauss, pos_pe, nap, sparse);
    k5_conv1<<<8192, 256, 0, stream>>>(diff, thresh, w1, b1, ga1, be1, x1);
    k6_conv2<<<2048, 256, 0, stream>>>(x1, w2, b2, ga2, be2, x2);
    k7_conv3_wmma<<<4096, 256, 0, stream>>>(x2, w3, b3, dense);
}